// QDense4StatesAncilla_26817775796832
// MI455X (gfx1250) — compile-verified
//
#include <hip/hip_runtime.h>
#include <math.h>

// CDNA5 / gfx1250: wave32, WMMA 16x16 shapes.
typedef __attribute__((ext_vector_type(2))) float v2f;
typedef __attribute__((ext_vector_type(8))) float v8f;

#define N_WIRES 11
#define DIM 2048          // 2^11 amplitudes
#define HALF 1024         // ancilla=0 amplitudes
#define NTHREADS 256      // 8 wave32 waves

__device__ __forceinline__ v8f wmma_f32(v2f a, v2f b, v8f c) {
  // V_WMMA_F32_16X16X4_F32 : D(16x16) = A(16x4) * B(4x16) + C
  return __builtin_amdgcn_wmma_f32_16x16x4_f32(
      false, a, false, b, (short)0, c, false, false);
}

// Composed CNOT-ring gather permutation (linear over GF(2)).
// Reference applies CNOT(q, (q+r)%11) for q = 0..10 in order; the gather
// index is sigma_0(sigma_1(...sigma_10(t)...)), i.e. apply q=10 first.
__device__ __forceinline__ unsigned perm_apply(unsigned t,
                                               const unsigned tmr[N_WIRES]) {
#pragma unroll
  for (int q = N_WIRES - 1; q >= 0; --q) {
    t ^= (t & (1u << (N_WIRES - 1 - q))) ? tmr[q] : 0u;  // wire q = bit 10-q
  }
  return t;
}

// Build fused 4-wire gate G[16][16] = kron(U_i0, U_i1, U_i2, U_i3).
__device__ __forceinline__ void build_group(float* g_re, float* g_im,
                                            const float (*uw_re)[4],
                                            const float (*uw_im)[4],
                                            int i0, int i1, int i2, int i3,
                                            int tid) {
  const int row = tid >> 4;
  const int col = tid & 15;
  const int widx[4] = {i0, i1, i2, i3};
  float pr = 1.f, pi = 0.f;
#pragma unroll
  for (int w = 0; w < 4; ++w) {
    const int rb = (row >> (3 - w)) & 1;   // group's first wire = MSB of k
    const int cb = (col >> (3 - w)) & 1;
    const float ur = uw_re[widx[w]][rb * 2 + cb];
    const float ui = uw_im[widx[w]][rb * 2 + cb];
    const float nr = pr * ur - pi * ui;
    const float ni = pr * ui + pi * ur;
    pr = nr; pi = ni;
  }
  g_re[tid] = pr;
  g_im[tid] = pi;
}

// Apply G (16x16 complex, LDS) to wires [gstart, gstart+3] as a complex GEMM
// G x V(16x128) via 16 chained V_WMMA_F32_16X16X4_F32 per wave.
// If perm_r >= 0, the deferred CNOT-ring permutation (range perm_r) is fused
// into the B-operand gather; src and dst must then be distinct buffers.
__device__ __forceinline__ void apply_group(const float* src_re,
                                            const float* src_im,
                                            float* dst_re, float* dst_im,
                                            const float* g_re,
                                            const float* g_im,
                                            int gstart, int perm_r, int tid) {
  const int lane = tid & 31;
  const int wave = tid >> 5;
  const int shift = N_WIRES - gstart - 4;   // log2(right)
  const int right = 1 << shift;

  // 128 columns total; each of the 8 waves owns 16 of them.
  const int col = wave * 16 + (lane & 15);
  const int li = col >> shift;
  const int ri = col & (right - 1);
  const int addr0 = li * (16 * right) + ri; // element k lives at addr0+k*right

  const int khalf = (lane >> 4) << 1;       // lanes 16-31 hold K+2,K+3
  const int M = lane & 15;

  // ---- read addresses (optionally routed through the deferred permutation;
  //      the permutation is GF(2)-linear so evaluate on basis vectors only) --
  unsigned eaddr[4], oaddr[4];
  if (perm_r >= 0) {
    unsigned tmr[N_WIRES];
#pragma unroll
    for (int q = 0; q < N_WIRES; ++q) {
      int s = q + perm_r;
      if (s >= N_WIRES) s -= N_WIRES;
      tmr[q] = 1u << (N_WIRES - 1 - s);     // target-bit mask of CNOT q
    }
    const unsigned a0 = perm_apply((unsigned)addr0, tmr);
    const unsigned p1 = perm_apply((unsigned)right, tmr);
    const unsigned p2 = perm_apply(2u * right, tmr);
    const unsigned p4 = perm_apply(4u * right, tmr);
    const unsigned p8 = perm_apply(8u * right, tmr);
    const unsigned aL = a0 ^ (khalf ? p2 : 0u);
#pragma unroll
    for (int kc = 0; kc < 4; ++kc) {
      unsigned e = aL;
      if (kc & 1) e ^= p4;
      if (kc & 2) e ^= p8;
      eaddr[kc] = e;
      oaddr[kc] = e ^ p1;
    }
  } else {
#pragma unroll
    for (int kc = 0; kc < 4; ++kc) {
      const int kk = kc * 4 + khalf;
      eaddr[kc] = (unsigned)(addr0 + kk * right);
      oaddr[kc] = eaddr[kc] + (unsigned)right;
    }
  }

  v2f aRe[4], aIm[4], aNegIm[4], bRe[4], bIm[4];
#pragma unroll
  for (int kc = 0; kc < 4; ++kc) {
    const int kk = kc * 4 + khalf;
    aRe[kc][0] = g_re[M * 16 + kk];
    aRe[kc][1] = g_re[M * 16 + kk + 1];
    aIm[kc][0] = g_im[M * 16 + kk];
    aIm[kc][1] = g_im[M * 16 + kk + 1];
    aNegIm[kc][0] = -aIm[kc][0];
    aNegIm[kc][1] = -aIm[kc][1];
    bRe[kc][0] = src_re[eaddr[kc]];
    bRe[kc][1] = src_re[oaddr[kc]];
    bIm[kc][0] = src_im[eaddr[kc]];
    bIm[kc][1] = src_im[oaddr[kc]];
  }

  v8f dRe = {0.f, 0.f, 0.f, 0.f, 0.f, 0.f, 0.f, 0.f};
  v8f dIm = {0.f, 0.f, 0.f, 0.f, 0.f, 0.f, 0.f, 0.f};
#pragma unroll
  for (int kc = 0; kc < 4; ++kc) dRe = wmma_f32(aRe[kc], bRe[kc], dRe);
#pragma unroll
  for (int kc = 0; kc < 4; ++kc) dRe = wmma_f32(aNegIm[kc], bIm[kc], dRe);
#pragma unroll
  for (int kc = 0; kc < 4; ++kc) dIm = wmma_f32(aRe[kc], bIm[kc], dIm);
#pragma unroll
  for (int kc = 0; kc < 4; ++kc) dIm = wmma_f32(aIm[kc], bRe[kc], dIm);

  // D layout: VGPR j holds row M = 8*(lane>>4)+j, column N = lane&15.
  // Writes are canonical (un-permuted); columns are disjoint across lanes.
  const int mbase = (lane >> 4) * 8;
#pragma unroll
  for (int j = 0; j < 8; ++j) {
    const int m = mbase + j;
    dst_re[addr0 + m * right] = dRe[j];
    dst_im[addr0 + m * right] = dIm[j];
  }
}

__global__ __launch_bounds__(NTHREADS)
void qsim_kernel(const float* __restrict__ x, const float* __restrict__ y,
                 const float* __restrict__ w0, const float* __restrict__ w1,
                 float* __restrict__ out) {
  __shared__ float sre[2][DIM];   // double-buffered state (re)
  __shared__ float sim[2][DIM];   // double-buffered state (im)
  __shared__ float g_re[3][256];  // three fused 16x16 gates per layer
  __shared__ float g_im[3][256];
  __shared__ float uw_re[13][4];  // 0..10: layer Rot gates, 11: I, 12: RX(y)
  __shared__ float uw_im[13][4];
  __shared__ float red[8];
  __shared__ float s_norm;

  const int tid = threadIdx.x;
  const int b = blockIdx.x;
  const int lane = tid & 31;
  const int wave = tid >> 5;

  // ---- coalesced image load (b128), wave-level norm reduction ----
  const float4 v4 = ((const float4*)(x + (size_t)b * HALF))[tid];
  float acc = v4.x * v4.x + v4.y * v4.y + v4.z * v4.z + v4.w * v4.w;
#pragma unroll
  for (int off = 16; off > 0; off >>= 1) acc += __shfl_xor(acc, off, 32);
  if (lane == 0) red[wave] = acc;
  __syncthreads();
  if (tid == 0) {
    float t = 0.f;
#pragma unroll
    for (int i = 0; i < 8; ++i) t += red[i];
    s_norm = sqrtf(t);
  }
  __syncthreads();
  const float norm = s_norm;
  const float inv = 1.0f / norm;

  // init state in buffer 0: amplitude 2k = x[k]/norm, ancilla=1 slots = 0
  {
    const int k4 = tid * 4;
    const float xv[4] = {v4.x, v4.y, v4.z, v4.w};
#pragma unroll
    for (int j = 0; j < 4; ++j) {
      const int k = k4 + j;
      sre[0][2 * k] = xv[j] * inv;
      sre[0][2 * k + 1] = 0.f;
      sim[0][2 * k] = 0.f;
      sim[0][2 * k + 1] = 0.f;
    }
  }
  if (tid == 0) {
    // slot 11: identity
    uw_re[11][0] = 1.f; uw_re[11][1] = 0.f; uw_re[11][2] = 0.f; uw_re[11][3] = 1.f;
    uw_im[11][0] = 0.f; uw_im[11][1] = 0.f; uw_im[11][2] = 0.f; uw_im[11][3] = 0.f;
    // slot 12: RX(y[b]) = [[c, -is],[-is, c]]
    float sh, ch;
    sincosf(y[b] * 0.5f, &sh, &ch);
    uw_re[12][0] = ch;  uw_re[12][1] = 0.f; uw_re[12][2] = 0.f; uw_re[12][3] = ch;
    uw_im[12][0] = 0.f; uw_im[12][1] = -sh; uw_im[12][2] = -sh; uw_im[12][3] = 0.f;
  }
  __syncthreads();

  int cur = 0;       // current state buffer (uniform)
  int pend_r = -1;   // deferred CNOT-ring range, consumed by next apply

  const float* wsets[2] = {w0, w1};
  for (int pass = 0; pass < 2; ++pass) {
    const float* w = wsets[pass];

    // RX on ancilla (wire 10) as fused gate I⊗I⊗I⊗RX on wires 7..10;
    // consumes any permutation deferred from the previous pass's last layer.
    build_group(g_re[0], g_im[0], uw_re, uw_im, 11, 11, 11, 12, tid);
    __syncthreads();
    if (pend_r >= 0) {
      apply_group(sre[cur], sim[cur], sre[cur ^ 1], sim[cur ^ 1],
                  g_re[0], g_im[0], 7, pend_r, tid);
      cur ^= 1;
    } else {
      apply_group(sre[cur], sim[cur], sre[cur], sim[cur],
                  g_re[0], g_im[0], 7, -1, tid);
    }
    pend_r = -1;
    __syncthreads();

    for (int l = 0; l < 6; ++l) {
      // per-layer Rot gates: Rot = RZ(om) RY(th) RZ(phi), angles = tanh(w)
      if (tid < N_WIRES) {
        const int base = (l * N_WIRES + tid) * 3;
        const float phi = tanhf(w[base + 0]);
        const float th  = tanhf(w[base + 1]);
        const float om  = tanhf(w[base + 2]);
        float sh, ch;
        sincosf(th * 0.5f, &sh, &ch);
        const float p = (phi + om) * 0.5f;
        const float m = (phi - om) * 0.5f;
        float sp, cp, sm, cm;
        sincosf(p, &sp, &cp);
        sincosf(m, &sm, &cm);
        uw_re[tid][0] = ch * cp;  uw_im[tid][0] = -ch * sp;  // a*c
        uw_re[tid][1] = -sh * cm; uw_im[tid][1] = -sh * sm;  // -b*s
        uw_re[tid][2] = sh * cm;  uw_im[tid][2] = -sh * sm;  // conj(b)*s
        uw_re[tid][3] = ch * cp;  uw_im[tid][3] = ch * sp;   // conj(a)*c
      }
      __syncthreads();

      // build all three fused gates in one phase (disjoint writes)
      build_group(g_re[0], g_im[0], uw_re, uw_im, 0, 1, 2, 3, tid);
      build_group(g_re[1], g_im[1], uw_re, uw_im, 4, 5, 6, 7, tid);
      build_group(g_re[2], g_im[2], uw_re, uw_im, 11, 8, 9, 10, tid);
      __syncthreads();

      // group 0 consumes the previous layer's deferred CNOT permutation
      if (pend_r >= 0) {
        apply_group(sre[cur], sim[cur], sre[cur ^ 1], sim[cur ^ 1],
                    g_re[0], g_im[0], 0, pend_r, tid);
        cur ^= 1;
      } else {
        apply_group(sre[cur], sim[cur], sre[cur], sim[cur],
                    g_re[0], g_im[0], 0, -1, tid);
      }
      pend_r = -1;
      __syncthreads();
      apply_group(sre[cur], sim[cur], sre[cur], sim[cur],
                  g_re[1], g_im[1], 4, -1, tid);
      __syncthreads();
      apply_group(sre[cur], sim[cur], sre[cur], sim[cur],
                  g_re[2], g_im[2], 7, -1, tid);
      __syncthreads();

      // defer this layer's CNOT ring (range r) as a lazy permutation
      pend_r = l % (N_WIRES - 1) + 1;
    }
  }

  // ---- output gather: consume final deferred permutation, scale, b128 store
  {
    unsigned tmr[N_WIRES];
#pragma unroll
    for (int q = 0; q < N_WIRES; ++q) {
      int s = q + pend_r;                 // pend_r = 6 here (layer 5 ring)
      if (s >= N_WIRES) s -= N_WIRES;
      tmr[q] = 1u << (N_WIRES - 1 - s);
    }
    const int k4 = tid * 4;
    float re[4], im[4];
#pragma unroll
    for (int j = 0; j < 4; ++j) {
      const unsigned idx = perm_apply((unsigned)(2 * (k4 + j)), tmr);
      re[j] = sre[cur][idx] * norm;
      im[j] = sim[cur][idx] * norm;
    }
    float4* o = (float4*)(out + (size_t)b * HALF * 2);
    o[tid * 2 + 0] = make_float4(re[0], im[0], re[1], im[1]);
    o[tid * 2 + 1] = make_float4(re[2], im[2], re[3], im[3]);
  }
}

extern "C" void kernel_launch(void* const* d_in, const int* in_sizes, int n_in,
                              void* d_out, int out_size, void* d_ws,
                              size_t ws_size, hipStream_t stream) {
  const float* x = (const float*)d_in[0];   // [B,1,32,32] float32
  const float* y = (const float*)d_in[1];   // [B] float32
  const float* w0 = (const float*)d_in[2];  // [6,11,3] float32
  const float* w1 = (const float*)d_in[3];  // [6,11,3] float32
  float* out = (float*)d_out;               // [B,1,32,32] complex64 (re,im)

  const int B = in_sizes[1];                // y has one angle per batch element
  qsim_kernel<<<B, NTHREADS, 0, stream>>>(x, y, w0, w1, out);
}